// VicregLoss_78993038508418
// MI455X (gfx1250) — compile-verified
//
#include <hip/hip_runtime.h>
#include <hip/hip_bf16.h>

// ---------------- problem constants ----------------
#define Bc 4
#define Zc 128
#define Pc 65536          // H*W
#define Lc 64             // labels per batch
#define SEGS (Bc * Lc)    // 256
#define TILE 64           // pixels per LDS tile (K-chunk)
#define XSTRIDE 68        // padded LDS row stride (floats): 16B aligned, conflict-free
#define TPB 256           // 8 waves of 32
#define BLOCKS_PER_BATCH 64
#define TILES_PER_BLOCK 16   // 64 blocks * 16 tiles * 64 px = 65536 = P

// workspace layout (floats)
#define WS_GA   0          // [128][128]
#define WS_GM   16384      // [128][128]
#define WS_NL   32768      // [256]
#define WS_SQA  33024      // [256]
#define WS_SQM  33280      // [256]
#define WS_SUMA 33536      // [256][128]
#define WS_SUMM 66304      // [256][128]
#define WS_FLOATS 99072

typedef float v2f __attribute__((ext_vector_type(2)));
typedef float v8f __attribute__((ext_vector_type(8)));
typedef unsigned int v4u __attribute__((ext_vector_type(4)));
typedef int v4i __attribute__((ext_vector_type(4)));
typedef int v8i __attribute__((ext_vector_type(8)));

// LDS layout (floats), dynamic shared. Double-buffered tiles for TDM pipelining.
#define SM_XT0   0        // [128][68]
#define SM_XT1   8704     // [128][68]
#define SM_SEG   17408    // [64][128]
#define SM_WTS   25600    // [64]
#define SM_SEGN  25664    // [64]
#define SM_SEGSQ 25728    // [64]
#define SM_LAB   25792    // [64] (int)
#define SM_FLOATS 25856   // 103,424 bytes (<< 320KB/WGP)

#if defined(__HIP_DEVICE_COMPILE__) && __has_builtin(__builtin_amdgcn_tensor_load_to_lds)
#define USE_TDM 1
#else
#define USE_TDM 0
#endif

__global__ void vicreg_zero(float* __restrict__ ws, int n) {
  int i = blockIdx.x * blockDim.x + threadIdx.x;
  if (i < n) ws[i] = 0.0f;
}

#if USE_TDM
// Issue one Tensor Data Mover DMA: 128 rows x 64 floats tile, global->LDS,
// row stride Pc floats in memory, LDS padding 4 DWORDs per 64 DWORDs (=> XSTRIDE 68).
// This toolchain's builtin is the 6-arg form:
//   (uint32x4 g0, int32x8 g1, int32x4 g2, int32x4 g3, int32x8 g4, i32 cpol)
__device__ __forceinline__ void tdm_load_tile(const float* gsrc, unsigned lds_byte_off) {
  const unsigned long long ga = (unsigned long long)gsrc;
  v4u g0;
  g0[0] = 1u;                                   // count=1, user descriptor
  g0[1] = lds_byte_off;                         // lds_addr (bytes)
  g0[2] = (unsigned)(ga & 0xFFFFFFFFu);         // global_addr[31:0]
  g0[3] = (unsigned)((ga >> 32) & 0x01FFFFFFu)  // global_addr[56:32]
          | (2u << 30);                         // type = 2 ("image")
  v8i g1;
  g1[0] = (int)((2u << 16)      // data_size = 4 bytes
              | (1u << 20)      // pad_enable
              | (5u << 22)      // pad_interval: 64 DWORDs
              | (3u << 25));    // pad_amount:   4 DWORDs  -> row stride 68 floats
  g1[1] = 0;                                    // atomic barrier addr, tensor_dim0 lo16=0
  g1[2] = (int)((Pc >> 16) | ((unsigned)Zc << 16)); // tensor_dim0 hi16=1, tensor_dim1 lo16=128
  g1[3] = (int)((unsigned)TILE << 16);          // tensor_dim1 hi16=0, tile_dim0 = 64
  g1[4] = Zc;                                   // tile_dim1 = 128, tile_dim2 = 0
  g1[5] = Pc;                                   // tensor_dim0_stride lo32 = 65536
  g1[6] = 0;                                    // stride hi16 = 0, tensor_dim1_stride lo16 = 0
  g1[7] = 0;
  const v4i gz4 = {0, 0, 0, 0};                 // groups 2/3 unused (2D tile)
  const v8i gz8 = {0, 0, 0, 0, 0, 0, 0, 0};
  __builtin_amdgcn_tensor_load_to_lds(g0, g1, gz4, gz4, gz8, 0);
}
#endif

__global__ void vicreg_main(const float* __restrict__ pred,
                            const float* __restrict__ predmv,
                            const int* __restrict__ gt,
                            const unsigned char* __restrict__ fov,
                            float* __restrict__ ws) {
  extern __shared__ float smem[];
  float* seg   = smem + SM_SEG;
  float* wts   = smem + SM_WTS;
  float* segN  = smem + SM_SEGN;
  float* segSq = smem + SM_SEGSQ;
  int*   lab   = (int*)(smem + SM_LAB);

  const int tid  = threadIdx.x;
  const int wave = tid >> 5;
  const int lane = tid & 31;
  const int m16  = lane & 15;     // row/col within 16x16 fragment
  const int kh   = lane >> 4;     // K half-select for f32 WMMA fragments
  const int rowbase = wave * 16;  // each wave owns 16 rows of G

  const int b        = blockIdx.x / BLOCKS_PER_BATCH;
  const int bib      = blockIdx.x % BLOCKS_PER_BATCH;
  const int pixBase  = bib * (TILES_PER_BLOCK * TILE);

  float* GA   = ws + WS_GA;
  float* GM   = ws + WS_GM;
  float* NL   = ws + WS_NL;
  float* SQA  = ws + WS_SQA;
  float* SQM  = ws + WS_SQM;
  float* SUMA = ws + WS_SUMA;
  float* SUMM = ws + WS_SUMM;

  for (int v = 0; v < 2; ++v) {
    const float* srcB = (v ? predmv : pred) + (size_t)b * Zc * Pc;
    float* SEGOUT = v ? SUMM : SUMA;
    float* SQOUT  = v ? SQM  : SQA;
    float* GOUT   = v ? GM   : GA;

    // zero per-view LDS accumulators
    for (int j = tid; j < Lc * Zc; j += TPB) seg[j] = 0.0f;
    if (tid < Lc) { segSq[tid] = 0.0f; if (v == 0) segN[tid] = 0.0f; }

    v8f acc[8];
    #pragma unroll
    for (int c = 0; c < 8; ++c)
      acc[c] = (v8f){0.f, 0.f, 0.f, 0.f, 0.f, 0.f, 0.f, 0.f};

#if USE_TDM
    // prime the pipeline: DMA tile 0 into buffer 0 (one wave issues; EXEC ignored by TDM)
    if (wave == 0)
      tdm_load_tile(srcB + pixBase, (unsigned)(SM_XT0 * sizeof(float)));
#endif

    for (int t = 0; t < TILES_PER_BLOCK; ++t) {
      const int cur  = t & 1;
      const int pix0 = pixBase + t * TILE;
      float* Xt = smem + (cur ? SM_XT1 : SM_XT0);

#if USE_TDM
      if (wave == 0) __builtin_amdgcn_s_wait_tensorcnt(0);  // tile t landed in LDS
      __syncthreads();  // broadcast completion; prev tile fully consumed
      // kick off DMA of tile t+1 into the other buffer while we compute on tile t
      if (t + 1 < TILES_PER_BLOCK && wave == 0)
        tdm_load_tile(srcB + pix0 + TILE,
                      (unsigned)((cur ? SM_XT0 : SM_XT1) * sizeof(float)));
      if (tid < TILE) {
        const size_t p = (size_t)b * Pc + pix0 + tid;
        const int g = gt[p];
        lab[tid] = g;
        wts[tid] = (fov[p] && g != 0) ? 1.0f : 0.0f;
      }
      __syncthreads();  // labels/weights visible
#else
      __syncthreads();  // previous tile fully consumed before restage
      if (tid < TILE) {
        const size_t p = (size_t)b * Pc + pix0 + tid;
        const int g = gt[p];
        lab[tid] = g;
        wts[tid] = (fov[p] && g != 0) ? 1.0f : 0.0f;
      }
      for (int j = tid; j < (Zc * TILE) / 4; j += TPB) {
        const int z  = j >> 4;
        const int c4 = j & 15;
        const float4 d = *(const float4*)(srcB + (size_t)z * Pc + pix0 + c4 * 4);
        *(float4*)(Xt + z * XSTRIDE + c4 * 4) = d;
      }
      if (t + 1 < TILES_PER_BLOCK && tid < Zc)
        __builtin_prefetch(srcB + (size_t)tid * Pc + pix0 + TILE, 0, 1);
      __syncthreads();
#endif

      // ---- Gram accumulation: G += (w .* X) * X^T via fp32 WMMA 16x16x4 ----
      // A fragment (16x4): lane L -> M = L%16, VGPR0/1 -> K = 2*(L/16) + {0,1}
      // B fragment (4x16): lane L -> N = L%16, VGPR0/1 -> K = 2*(L/16) + {0,1}
      // Batch the 9 LDS fragment loads per K-step before the 8 WMMAs so the
      // scheduler can pipeline ds traffic against the matrix pipe.
      #pragma unroll
      for (int k0 = 0; k0 < TILE; k0 += 4) {
        const int ka = k0 + 2 * kh;
        v2f a;
        a.x = Xt[(rowbase + m16) * XSTRIDE + ka];
        a.y = Xt[(rowbase + m16) * XSTRIDE + ka + 1];
        v2f fb[8];
        #pragma unroll
        for (int c = 0; c < 8; ++c) {
          fb[c].x = Xt[(c * 16 + m16) * XSTRIDE + ka];
          fb[c].y = Xt[(c * 16 + m16) * XSTRIDE + ka + 1];
        }
        a.x *= wts[ka];
        a.y *= wts[ka + 1];
        #pragma unroll
        for (int c = 0; c < 8; ++c)
          acc[c] = __builtin_amdgcn_wmma_f32_16x16x4_f32(
              false, a, false, fb[c], (short)0, acc[c], false, false);
      }

      // ---- per-(label, feature) segment sums via LDS atomics ----
      {
        const int z = tid & 127;
        for (int i = tid >> 7; i < TILE; i += 2) {
          const float w = wts[i];
          if (w != 0.0f)
            atomicAdd(&seg[(lab[i] << 7) + z], Xt[z * XSTRIDE + i]);
        }
      }
      // per-label counts and squared norms (one thread per pixel)
      if (tid < TILE) {
        const float w = wts[tid];
        if (w != 0.0f) {
          if (v == 0) atomicAdd(&segN[lab[tid]], 1.0f);
          float s = 0.0f;
          for (int z = 0; z < Zc; ++z) {
            const float x = Xt[z * XSTRIDE + tid];
            s += x * x;
          }
          atomicAdd(&segSq[lab[tid]], s);
        }
      }
    } // tiles

    __syncthreads();
    // flush segment sums to global
    for (int j = tid; j < Lc * Zc; j += TPB)
      atomicAdd(&SEGOUT[b * (Lc * Zc) + j], seg[j]);
    if (tid < Lc) {
      atomicAdd(&SQOUT[b * Lc + tid], segSq[tid]);
      if (v == 0) atomicAdd(&NL[b * Lc + tid], segN[tid]);
    }
    // flush Gram slab: C/D layout — VGPR r: lanes 0-15 -> M=r, lanes 16-31 -> M=r+8
    #pragma unroll
    for (int c = 0; c < 8; ++c) {
      #pragma unroll
      for (int r = 0; r < 8; ++r) {
        const int row = rowbase + r + (kh ? 8 : 0);
        const int col = c * 16 + m16;
        atomicAdd(&GOUT[row * Zc + col], acc[c][r]);
      }
    }
    __syncthreads();
  } // views
}

__device__ __forceinline__ float block_reduce_sum(float* red, float val, int tid) {
  red[tid] = val;
  __syncthreads();
  for (int s = 128; s > 0; s >>= 1) {
    if (tid < s) red[tid] += red[tid + s];
    __syncthreads();
  }
  const float r = red[0];
  __syncthreads();
  return r;
}

__global__ void vicreg_finalize(const float* __restrict__ ws, float* __restrict__ out) {
  __shared__ float red[TPB];
  __shared__ float meanA[Zc];
  __shared__ float meanM[Zc];

  const float* GA   = ws + WS_GA;
  const float* GM   = ws + WS_GM;
  const float* NL   = ws + WS_NL;
  const float* SQA  = ws + WS_SQA;
  const float* SQM  = ws + WS_SQM;
  const float* SUMA = ws + WS_SUMA;
  const float* SUMM = ws + WS_SUMM;
  const int tid = threadIdx.x;

  // ---- sim loss pieces ----
  float pn = 0.f, ptotal = 0.f, pcount = 0.f;
  for (int s = tid; s < SEGS; s += TPB) {
    const float nl = NL[s];
    pn     += nl;
    ptotal += nl * (SQA[s] + SQM[s]);
    pcount += nl * nl;
  }
  float pdot = 0.f;
  for (int j = tid; j < SEGS * Zc; j += TPB) pdot += SUMA[j] * SUMM[j];

  const float n     = block_reduce_sum(red, pn, tid);
  const float total = block_reduce_sum(red, ptotal, tid);
  const float count = block_reduce_sum(red, pcount, tid);
  const float dot   = block_reduce_sum(red, pdot, tid);
  const float sim_loss = (total - 2.0f * dot) / count;

  // ---- global masked means per feature (column sums over all segments) ----
  if (tid < Zc) {
    float sa = 0.f, sm = 0.f;
    for (int s = 0; s < SEGS; ++s) {
      sa += SUMA[s * Zc + tid];
      sm += SUMM[s * Zc + tid];
    }
    meanA[tid] = sa / n;
    meanM[tid] = sm / n;
  }
  __syncthreads();

  // ---- std loss from Gram diagonal ----
  float pstd = 0.f;
  if (tid < Zc) {
    const float va = (GA[tid * Zc + tid] - n * meanA[tid] * meanA[tid]) / (n - 1.0f);
    const float vm = (GM[tid * Zc + tid] - n * meanM[tid] * meanM[tid]) / (n - 1.0f);
    pstd = fmaxf(0.f, 1.f - sqrtf(va + 1e-4f)) + fmaxf(0.f, 1.f - sqrtf(vm + 1e-4f));
  }
  const float std_loss = block_reduce_sum(red, pstd, tid) / (float)Zc;

  // ---- cov loss: off-diagonal of (G - n mean mean^T)/(n-1), squared ----
  float pcov = 0.f;
  for (int j = tid; j < Zc * Zc; j += TPB) {
    const int z = j >> 7, k = j & 127;
    if (z != k) {
      const float ca = (GA[j] - n * meanA[z] * meanA[k]) / (n - 1.0f);
      const float cm = (GM[j] - n * meanM[z] * meanM[k]) / (n - 1.0f);
      pcov += ca * ca + cm * cm;
    }
  }
  const float cov_loss = block_reduce_sum(red, pcov, tid) / (float)Zc;

  if (tid == 0) out[0] = sim_loss + std_loss + cov_loss;
}

extern "C" void kernel_launch(void* const* d_in, const int* in_sizes, int n_in,
                              void* d_out, int out_size, void* d_ws, size_t ws_size,
                              hipStream_t stream) {
  const float* pred          = (const float*)d_in[0];
  const float* predmv        = (const float*)d_in[1];
  const int* gt              = (const int*)d_in[2];
  const unsigned char* fov   = (const unsigned char*)d_in[3];
  float* ws  = (float*)d_ws;
  float* out = (float*)d_out;

  vicreg_zero<<<(WS_FLOATS + TPB - 1) / TPB, TPB, 0, stream>>>(ws, WS_FLOATS);

  const size_t shmem = (size_t)SM_FLOATS * sizeof(float);  // ~101 KB, fine on 320KB WGP LDS
  vicreg_main<<<Bc * BLOCKS_PER_BATCH, TPB, shmem, stream>>>(pred, predmv, gt, fov, ws);

  vicreg_finalize<<<1, TPB, 0, stream>>>(ws, out);
}